// DecompModel_4114578669580
// MI455X (gfx1250) — compile-verified
//
#include <hip/hip_runtime.h>
#include <hip/hip_bf16.h>
#include <stdint.h>

// Problem constants (from reference): B=16, T=144 (+1 empty), C=1, H=W=128, L=6
#define NB    16
#define NT    144
#define NTP1  145
#define NPIX  16384          // 128*128
#define NSEL  6
#define TPB   256            // 8 wave32 per block
#define NTMPL 4              // templates evaluated per error block
#define NGRP  (NT / NTMPL)   // 36 real-template groups (+1 empty block)
#define TILE  1024           // pixels per streamed tile
#define NTILES (NPIX / TILE) // 16
#define PVEC   (TILE / TPB)  // 4 floats per thread per tile (1 x b128 per array)

typedef __attribute__((ext_vector_type(2))) float v2f;
typedef __attribute__((ext_vector_type(8))) float v8f;

// ---------------------------------------------------------------------------
// CDNA5 async global->LDS streaming (ASYNCcnt path).  Templates use th:NT so
// they do not evict the (L2-resident, 151MB < 192MB) mask array; masks use the
// default RT policy so they stay hot in L2 across all 6 greedy steps.
// ---------------------------------------------------------------------------
__device__ __forceinline__ void async_ld16_nt(const float* g, unsigned lds) {
    asm volatile("global_load_async_to_lds_b128 %0, %1, off th:TH_LOAD_NT"
                 :: "v"(lds), "v"(g) : "memory");
}
__device__ __forceinline__ void async_ld16_rt(const float* g, unsigned lds) {
    asm volatile("global_load_async_to_lds_b128 %0, %1, off"
                 :: "v"(lds), "v"(g) : "memory");
}
__device__ __forceinline__ void wait_async_le8() {
    asm volatile("s_wait_asynccnt 8" ::: "memory");
}
__device__ __forceinline__ void wait_async_0() {
    asm volatile("s_wait_asynccnt 0" ::: "memory");
}

// ---------------------------------------------------------------------------
// Exact f32 wave reduction using V_WMMA_F32_16X16X4_F32:
//   A = ones(16x4), B VGPR0 = per-lane partial, B VGPR1 = 0
//   => every row of D = column sums s_n = v_n + v_{n+16}; lane n (and n+16)
//      of d[0] holds s_{n&15}.  Then 4 ds_swizzle XOR folds (SWAPX8/4/2/1)
//      finish the 16-way sum.  f32 in / f32 accumulate -> exact up to order.
// ---------------------------------------------------------------------------
__device__ __forceinline__ float wave_sum_wmma(float acc) {
    v2f a;  a[0] = 1.0f; a[1] = 1.0f;
    v2f bm; bm[0] = acc; bm[1] = 0.0f;
    v8f c = {0.f, 0.f, 0.f, 0.f, 0.f, 0.f, 0.f, 0.f};
    v8f d = __builtin_amdgcn_wmma_f32_16x16x4_f32(false, a, false, bm,
                                                  (short)0, c, false, false);
    float s = d[0];
    s += __int_as_float(__builtin_amdgcn_ds_swizzle(__float_as_int(s), 0x201F)); // SWAPX8
    s += __int_as_float(__builtin_amdgcn_ds_swizzle(__float_as_int(s), 0x101F)); // SWAPX4
    s += __int_as_float(__builtin_amdgcn_ds_swizzle(__float_as_int(s), 0x081F)); // SWAPX2
    s += __int_as_float(__builtin_amdgcn_ds_swizzle(__float_as_int(s), 0x041F)); // SWAPX1
    return s;   // full wave sum in every lane
}

// ---------------------------------------------------------------------------
// init: A=0, Mprod=1, e0 = x - bg, zero the 'used' table.
// ---------------------------------------------------------------------------
__global__ __launch_bounds__(TPB) void init_kernel(
    const float* __restrict__ x, const float* __restrict__ bg,
    float* __restrict__ A, float* __restrict__ M, float* __restrict__ e0,
    int* __restrict__ used)
{
    int g = blockIdx.x * TPB + threadIdx.x;     // 0 .. NB*NPIX-1
    int p = g & (NPIX - 1);
    A[g]  = 0.0f;
    M[g]  = 1.0f;
    e0[g] = x[g] - bg[p];
    if (g < NB * NTP1) used[g] = 0;
}

// ---------------------------------------------------------------------------
// Error kernel: blockIdx.x==0 handles the empty template (t=0); block g>=1
// handles templates 4(g-1)+1 .. 4(g-1)+4 for batch blockIdx.y.
//   err[b][t] = sum_p r^2,  r = e0 - M*m*(t - bg)   (empty: m=1, t=5)
// Template/mask tiles are double-buffer async-streamed into LDS; per-batch
// fields e0/M/bg are loaded once per 4-pixel group and reused across the 4
// templates (4x less L2 traffic than one-template-per-block).
// ---------------------------------------------------------------------------
__global__ __launch_bounds__(TPB) void err_kernel(
    const float* __restrict__ templates, const float* __restrict__ masks,
    const float* __restrict__ bg, const float* __restrict__ e0,
    const float* __restrict__ M, float* __restrict__ err)
{
    const int g   = blockIdx.x;    // 0 = empty, 1..36 = template groups
    const int b   = blockIdx.y;    // 0..15
    const int tid = threadIdx.x;

    __shared__ float ldsT[2][NTMPL][TILE];   // 32 KB
    __shared__ float ldsM[2][NTMPL][TILE];   // 32 KB
    __shared__ float wsum[TPB / 32][NTMPL];

    const float* e0b = e0 + b * NPIX;
    const float* Mb  = M  + b * NPIX;

    float acc[NTMPL] = {0.0f, 0.0f, 0.0f, 0.0f};

    if (g == 0) {
        // empty template: value 5, mask 1  ->  r = e0 - M*(5 - bg)
        for (int p0 = tid * 4; p0 < NPIX; p0 += TPB * 4) {
            float4 e4 = *(const float4*)(e0b + p0);
            float4 M4 = *(const float4*)(Mb  + p0);
            float4 g4 = *(const float4*)(bg  + p0);
            float r;
            r = fmaf(-M4.x, 5.0f - g4.x, e4.x); acc[0] = fmaf(r, r, acc[0]);
            r = fmaf(-M4.y, 5.0f - g4.y, e4.y); acc[0] = fmaf(r, r, acc[0]);
            r = fmaf(-M4.z, 5.0f - g4.z, e4.z); acc[0] = fmaf(r, r, acc[0]);
            r = fmaf(-M4.w, 5.0f - g4.w, e4.w); acc[0] = fmaf(r, r, acc[0]);
        }
    } else {
        const int t0 = (g - 1) * NTMPL;                 // 0-based template idx
        const size_t tb = ((size_t)b * NT + t0) * NPIX;
        const float* tp = templates + tb;
        const float* mp = masks + tb;
        const unsigned ldsT0 = (unsigned)(uintptr_t)(&ldsT[0][0][0]);
        const unsigned ldsM0 = (unsigned)(uintptr_t)(&ldsM[0][0][0]);

        // one tile: 1 b128 template load (NT) + 1 b128 mask load (RT) per
        // thread per template => 8 ASYNCcnt increments per wave per tile.
        auto issue = [&](int tile, int buf) {
#pragma unroll
            for (int j = 0; j < NTMPL; ++j) {
                const float* gt = tp + (size_t)j * NPIX + tile * TILE + tid * PVEC;
                const float* gm = mp + (size_t)j * NPIX + tile * TILE + tid * PVEC;
                unsigned lt = ldsT0 + (unsigned)((buf * NTMPL + j) * TILE + tid * PVEC) * 4u;
                unsigned lm = ldsM0 + (unsigned)((buf * NTMPL + j) * TILE + tid * PVEC) * 4u;
                async_ld16_nt(gt, lt);
                async_ld16_rt(gm, lm);
            }
        };

        issue(0, 0);
        for (int it = 0; it < NTILES; ++it) {
            if (it + 1 < NTILES) { issue(it + 1, (it + 1) & 1); wait_async_le8(); }
            else                 { wait_async_0(); }
            __syncthreads();                 // tile 'it' fully in LDS (all waves)
            const int buf  = it & 1;
            const int pix  = tid * PVEC;     // offset within tile
            const int base = it * TILE + pix;
            float4 e4 = *(const float4*)(e0b + base);
            float4 M4 = *(const float4*)(Mb  + base);
            float4 g4 = *(const float4*)(bg  + base);
#pragma unroll
            for (int j = 0; j < NTMPL; ++j) {
                float4 t4 = *(const float4*)(&ldsT[buf][j][pix]);
                float4 m4 = *(const float4*)(&ldsM[buf][j][pix]);
                float r;
                r = fmaf(-M4.x, m4.x * (t4.x - g4.x), e4.x); acc[j] = fmaf(r, r, acc[j]);
                r = fmaf(-M4.y, m4.y * (t4.y - g4.y), e4.y); acc[j] = fmaf(r, r, acc[j]);
                r = fmaf(-M4.z, m4.z * (t4.z - g4.z), e4.z); acc[j] = fmaf(r, r, acc[j]);
                r = fmaf(-M4.w, m4.w * (t4.w - g4.w), e4.w); acc[j] = fmaf(r, r, acc[j]);
            }
            __syncthreads();   // protect buffer about to be overwritten next iter
        }
    }

    // block reduction: per-accumulator WMMA wave fold + tiny LDS combine
#pragma unroll
    for (int j = 0; j < NTMPL; ++j) {
        float s = wave_sum_wmma(acc[j]);
        if ((tid & 31) == 0) wsum[tid >> 5][j] = s;
    }
    __syncthreads();
    if (g == 0) {
        if (tid == 0) {
            float tot = 0.0f;
#pragma unroll
            for (int w = 0; w < TPB / 32; ++w) tot += wsum[w][0];
            err[b * NTP1] = tot;             // unscaled SSE (argmin-equivalent)
        }
    } else if (tid < NTMPL) {
        float tot = 0.0f;
#pragma unroll
        for (int w = 0; w < TPB / 32; ++w) tot += wsum[w][tid];
        err[b * NTP1 + 1 + (g - 1) * NTMPL + tid] = tot;
    }
}

// ---------------------------------------------------------------------------
// Select: per-batch argmin over 145 candidates with 'used' masking
// (first-index tie-break, matching jnp.argmin).  16 lanes, one per batch.
// ---------------------------------------------------------------------------
__global__ void select_kernel(const float* __restrict__ err,
                              int* __restrict__ used, int* __restrict__ selid,
                              float* __restrict__ ids_out, int l)
{
    int b = threadIdx.x;
    if (b >= NB) return;
    const float* e = err + b * NTP1;
    const int*   u = used + b * NTP1;
    float best = 3.0e38f;
    int   bi   = 0;
    for (int t = 0; t < NTP1; ++t) {
        float v = u[t] ? 1.0e8f : e[t];   // BIG mask (SSE << 1e8 always)
        if (v < best) { best = v; bi = t; }
    }
    selid[b] = bi;
    if (bi != 0) used[b * NTP1 + bi] = 1;
    ids_out[b * NSEL + l] = (float)bi;    // ids emitted as float into d_out
}

// ---------------------------------------------------------------------------
// Update: gather selected template/mask, update A, Mprod, recompute e0 for
// the next step, and emit objects/omasks for this step straight into d_out.
// ---------------------------------------------------------------------------
__global__ __launch_bounds__(TPB) void update_kernel(
    const float* __restrict__ x, const float* __restrict__ templates,
    const float* __restrict__ masks, const float* __restrict__ bg,
    float* __restrict__ A, float* __restrict__ M, float* __restrict__ e0,
    const int* __restrict__ selid,
    float* __restrict__ objects, float* __restrict__ omasks, int l)
{
    int g = blockIdx.x * TPB + threadIdx.x;     // 0 .. NB*NPIX-1
    int b = g >> 14;
    int p = g & (NPIX - 1);
    int sel = selid[b];
    float tv, mv;
    if (sel == 0) { tv = 5.0f; mv = 1.0f; }
    else {
        size_t base = ((size_t)b * NT + (sel - 1)) * NPIX + p;
        tv = templates[base];
        mv = masks[base];
    }
    float Ao = A[g], Mo = M[g];
    float An = fmaf(Mo * tv, mv, Ao);     // A + M*t*m
    float Mn = Mo * (1.0f - mv);
    A[g] = An;
    M[g] = Mn;
    e0[g] = x[g] - An - Mn * bg[p];
    size_t o = ((size_t)b * NSEL + l) * NPIX + p;
    objects[o] = tv;
    omasks[o]  = mv;
}

// ---------------------------------------------------------------------------
// Compose: paint selected objects last->first over background
// (reference paints index 5 first = bottom, index 0 last = top).
// ---------------------------------------------------------------------------
__global__ __launch_bounds__(TPB) void compose_kernel(
    const float* __restrict__ bg, const float* __restrict__ objects,
    const float* __restrict__ omasks, float* __restrict__ recons)
{
    int g = blockIdx.x * TPB + threadIdx.x;
    int b = g >> 14;
    int p = g & (NPIX - 1);
    float r = bg[p];
#pragma unroll
    for (int i = NSEL - 1; i >= 0; --i) {
        size_t o = ((size_t)b * NSEL + i) * NPIX + p;
        float ov = objects[o];
        float mv = omasks[o];
        r = fmaf(ov, mv, r * (1.0f - mv));
    }
    recons[g] = r;
}

// ---------------------------------------------------------------------------
// Host launcher.  Inputs: x(262144), templates(37748736), masks(37748736),
// background(16384), all f32.  Output (f32, flat): recons | objects | omasks |
// ids.  Workspace use: ~3.2 MB (e0, A, Mprod, err, used, selid).
// ---------------------------------------------------------------------------
extern "C" void kernel_launch(void* const* d_in, const int* in_sizes, int n_in,
                              void* d_out, int out_size, void* d_ws, size_t ws_size,
                              hipStream_t stream) {
    const float* x         = (const float*)d_in[0];
    const float* templates = (const float*)d_in[1];
    const float* masks     = (const float*)d_in[2];
    const float* bg        = (const float*)d_in[3];

    float* ws    = (float*)d_ws;
    float* e0    = ws;                          // NB*NPIX
    float* A     = ws + NB * NPIX;              // NB*NPIX
    float* M     = ws + 2 * NB * NPIX;          // NB*NPIX
    float* err   = ws + 3 * NB * NPIX;          // NB*NTP1
    int*   used  = (int*)(err + NB * NTP1);     // NB*NTP1
    int*   selid = (int*)(err + 2 * NB * NTP1); // NB

    float* out     = (float*)d_out;
    float* recons  = out;                                   // 262144
    float* objects = out + NB * NPIX;                       // 1572864
    float* omasks  = objects + NB * NSEL * NPIX;            // 1572864
    float* idsf    = omasks + NB * NSEL * NPIX;             // 96

    const int pixBlocks = (NB * NPIX) / TPB;                // 1024

    init_kernel<<<pixBlocks, TPB, 0, stream>>>(x, bg, A, M, e0, used);

    for (int l = 0; l < NSEL; ++l) {
        err_kernel<<<dim3(NGRP + 1, NB), TPB, 0, stream>>>(templates, masks, bg,
                                                           e0, M, err);
        select_kernel<<<1, 32, 0, stream>>>(err, used, selid, idsf, l);
        update_kernel<<<pixBlocks, TPB, 0, stream>>>(x, templates, masks, bg,
                                                     A, M, e0, selid,
                                                     objects, omasks, l);
    }

    compose_kernel<<<pixBlocks, TPB, 0, stream>>>(bg, objects, omasks, recons);
}